// kernel_generated_0_62904091017417
// MI455X (gfx1250) — compile-verified
//
#include <hip/hip_runtime.h>

typedef float v2f __attribute__((ext_vector_type(2)));
typedef float v4f __attribute__((ext_vector_type(4)));
typedef float v8f __attribute__((ext_vector_type(8)));

namespace {
constexpr int WIMG     = 112;
constexpr int KTOT     = 84;     // 12 channels * 7 taps
constexpr int KSTEPS   = 21;     // 84 / 4
constexpr int XMSTRIDE = 128;    // padded row stride of im2col panel
}

// Implicit-GEMM grouped conv via V_WMMA_F32_16X16X4_F32.
// Block = 224 threads (7 waves) handles one (n, g, h) image row.
// Phase 1: build im2col panel Xm[84][128] + A fragments in LDS (div-free).
// Phase 2: each wave runs 21 WMMAs; K loop is DS+WMMA only (pure-immediate offsets).
__global__ __launch_bounds__(224) void conv1d_grouped_wmma_f32(
    const float* __restrict__ x,      // (128, 24, 112, 112)
    const float* __restrict__ wgt,    // (12, 12, 7)  [i][o][t]
    float* __restrict__ out)          // (128, 24, 112, 112)
{
    __shared__ float xm[KTOT][XMSTRIDE];      // B panel: Xm[k][col], 43 KB
    __shared__ v2f   wf[KSTEPS * 2][16];      // A pairs (Wm[2kk][m], Wm[2kk+1][m])

    const int tid = threadIdx.x;
    const int bid = blockIdx.x;
    const int h   = bid % WIMG;
    const int g   = (bid / WIMG) & 1;
    const int n   = bid / (2 * WIMG);

    // ---- stage weights: Wm[o][k] = wgt[i][o][t], k = i*7 + t, o<12 (pad 16) ----
    for (int idx = tid; idx < KSTEPS * 2 * 16; idx += 224) {
        const int kk = idx >> 4;              // 0..41
        const int m  = idx & 15;
        const unsigned k0 = 2u * kk, k1 = k0 + 1u;
        const unsigned i0 = k0 / 7u, t0 = k0 - 7u * i0;
        const unsigned i1 = k1 / 7u, t1 = k1 - 7u * i1;
        v2f a;
        a.x = (m < 12) ? wgt[i0 * 84 + m * 7 + t0] : 0.f;
        a.y = (m < 12) ? wgt[i1 * 84 + m * 7 + t1] : 0.f;
        wf[kk][m] = a;
    }

    // ---- seed halo zeros as b128: cols {0..3} and {100..111} of every k-row ----
    // (superset of the true halo; interior part is overwritten by the scatter)
    for (int idx = tid; idx < KTOT * 4; idx += 224) {
        const int row = idx >> 2;
        const int q   = idx & 3;
        const int col = q ? (96 + 4 * q) : 0;   // 0, 100, 104, 108
        *(v4f*)&xm[row][col] = (v4f){0.f, 0.f, 0.f, 0.f};
    }
    __syncthreads();   // halo zeros must land before valid-value scatter

    // ---- im2col scatter: x[ch][wc] -> Xm[ch*7+t][wc+3-t], div-free ----
    // blockDim = 224 = 2*112, so wc is invariant and ch steps by 2.
    const int ch0 = (tid >= WIMG) ? 1 : 0;
    const int wc  = tid - ch0 * WIMG;
    const float* xcol = x + (((size_t)(n * 24 + g * 12)) * WIMG + h) * WIMG + wc;
#pragma unroll
    for (int it = 0; it < 6; ++it) {
        const int ch = ch0 + 2 * it;
        const float v = xcol[(size_t)ch * WIMG * WIMG];
#pragma unroll
        for (int t = 0; t < 7; ++t) {
            const int col = wc + 3 - t;           // guards depend only on (wc,t): CSE'd
            if (col >= 0 && col < WIMG)
                xm[ch * 7 + t][col] = v;
        }
    }
    __syncthreads();

    // ---- WMMA phase: D[16x16] tile per wave, K loop is DS+WMMA only ----
    const int lane = tid & 31;
    const int wave = tid >> 5;                // w-chunk 0..6
    const int m    = lane & 15;               // A row / B col
    const int hl   = lane >> 4;               // half-wave K-pair select
    const int w0   = wave * 16;

    v8f acc = {};
#pragma unroll
    for (int j = 0; j < KSTEPS; ++j) {
        v2f a = wf[2 * j + hl][m];            // base + 256*j
        v2f b;
        b.x = xm[4 * j + 2 * hl][w0 + m];     // base + 2048*j
        b.y = xm[4 * j + 2 * hl + 1][w0 + m]; // base + 2048*j + 512
        acc = __builtin_amdgcn_wmma_f32_16x16x4_f32(
            /*neg_a=*/false, a, /*neg_b=*/false, b,
            /*c_mod=*/(short)0, acc, /*reuse_a=*/false, /*reuse_b=*/false);
    }

    // ---- store: VGPR r holds row (r + 8*hl); fold roll(+1, axis=H) in ----
    const int h_out = (h == WIMG - 1) ? 0 : (h + 1);
    float* orow = out + (((size_t)(n * 24 + g * 12)) * WIMG + h_out) * WIMG + w0 + m;
#pragma unroll
    for (int r = 0; r < 8; ++r) {
        const int mrow = r + 8 * hl;
        if (mrow < 12) {
            orow[(size_t)mrow * WIMG * WIMG] = acc[r];
        }
    }
}

extern "C" void kernel_launch(void* const* d_in, const int* in_sizes, int n_in,
                              void* d_out, int out_size, void* d_ws, size_t ws_size,
                              hipStream_t stream) {
    (void)in_sizes; (void)n_in; (void)out_size; (void)d_ws; (void)ws_size;
    const float* x   = (const float*)d_in[0];   // 128*24*112*112 fp32
    const float* wgt = (const float*)d_in[1];   // 12*12*7 fp32
    float* out       = (float*)d_out;

    const int blocks = 128 * 2 * WIMG;          // (n, g, h)
    conv1d_grouped_wmma_f32<<<blocks, 224, 0, stream>>>(x, wgt, out);
}